// CapsuleNeuralNetwork_51299089383479
// MI455X (gfx1250) — compile-verified
//
#include <hip/hip_runtime.h>

typedef float v2f __attribute__((ext_vector_type(2)));
typedef float v8f __attribute__((ext_vector_type(8)));
typedef unsigned int v4u __attribute__((ext_vector_type(4)));
typedef int v4i __attribute__((ext_vector_type(4)));
typedef int v8i __attribute__((ext_vector_type(8)));

#define TM 16            // batch tile = WMMA M
#define NT 8             // tall capsules
#define PS 1024          // primary slice per capsule row
#define F  512           // feature size
#define K0 1536          // layer0 K (1024 prim + 512 feat); col folded into bias
#define K1 512           // layer1 K (512 feat); col folded into bias
#define NCOL 4
#define NBATCH 4096
#define SROW 4100        // state row stride (4096 + 4 pad -> conflict-free banks)
#define QROW 260         // prim quarter-stage row stride (256 + 4 pad, TDM pad)
#define LDS_P (TM*SROW)  // 65600 floats = 262400 B
#define QBUF  (TM*QROW)  //  4160 floats =  16640 B  (x2 = 33280 B; total 295680 B)

#define HAVE_TDM __has_builtin(__builtin_amdgcn_tensor_load_to_lds)

static __device__ __forceinline__ v8f wmma4(v2f a, v2f b, v8f c) {
  // native fp32 WMMA: D(16x16) += A(16x4) x B(4x16), full f32 precision
  return __builtin_amdgcn_wmma_f32_16x16x4_f32(false, a, false, b, (short)0, c,
                                               false, false);
}

#if HAVE_TDM
// Async DMA one 16x256-float tile of prim into LDS (row stride 260 floats via
// TDM pad: 4 dwords after every 256 dwords). Issued by one wave; tracked with
// TENSORcnt.
static __device__ __forceinline__ void tdm_prim_tile(const float* prim,
                                                     long elem_off,
                                                     unsigned ldsoff) {
  unsigned long long ga =
      (unsigned long long)(const void*)prim + (unsigned long long)elem_off * 4ull;
  v4u g0;
  g0.x = 1u;                                        // count=1, user mode
  g0.y = ldsoff;                                    // lds_addr (bytes)
  g0.z = (unsigned)(ga & 0xffffffffu);              // global_addr[31:0]
  g0.w = (unsigned)((ga >> 32) & 0x01ffffffu) | 0x80000000u; // addr[56:32]|type=2
  v8i g1;
  // data_size=2(4B) | pad_enable | pad_interval=7(256dw) | pad_amount=3(4dw)
  g1[0] = (int)((2u << 16) | (1u << 20) | (7u << 22) | (3u << 25));
  g1[1] = (int)(256u << 16);                        // tensor_dim0 = 256
  g1[2] = (int)(16u << 16);                         // tensor_dim1 = 16
  g1[3] = (int)(256u << 16);                        // tile_dim0 = 256
  g1[4] = 16;                                       // tile_dim1 = 16
  g1[5] = 8192;                                     // tensor_dim0_stride
  g1[6] = 0;
  g1[7] = 0;
  v4i gz4 = {0, 0, 0, 0};                           // groups 2/3: unused (2D)
  v8i gz8 = {0, 0, 0, 0, 0, 0, 0, 0};
  __builtin_amdgcn_tensor_load_to_lds(g0, g1, gz4, gz4, gz8, 0);
}
#endif

__global__ __launch_bounds__(256)
void caps_main(const float* __restrict__ prim,
               const float* __restrict__ Wt0, const float* __restrict__ Wt1,
               const float* __restrict__ eb0, const float* __restrict__ eb1,
               const float* __restrict__ Wout, const float* __restrict__ bout,
               float* __restrict__ out)
{
  __shared__ float lds[LDS_P + 2*QBUF];
  float* P  = lds;                 // rotating state, 16 x 4096 (stride 4100)
  float* Qb[2] = { lds + LDS_P, lds + LDS_P + QBUF };  // prim quarter tiles

  const int tid  = threadIdx.x;
  const int wave = tid >> 5;           // 0..7
  const int lane = tid & 31;
  const int mr   = lane & 15;          // M (batch row) for A / N column for B
  const int kh   = (lane >> 4) << 1;   // K pair select: 0 or 2
  const int mhi  = (lane >> 4) << 3;   // C-layout M offset: 0 or 8
  const long b0r = (long)blockIdx.x * TM;

  for (int i = tid; i < LDS_P; i += 256) P[i] = 0.f;
  __syncthreads();

  const v8f vzero = {0.f,0.f,0.f,0.f,0.f,0.f,0.f,0.f};
  int off = 0;  // logical rotation of the state (implements jnp.roll for free)

  for (int col = 0; col < NCOL; ++col) {
    // ---------------- layer 0: K = 1024 (prim via TDM) + 512 (state) --------
    for (int t = 0; t < NT; ++t) {
      v8f acc[4] = {vzero, vzero, vzero, vzero};
      const float* bw[4];
      #pragma unroll
      for (int j = 0; j < 4; ++j)
        bw[j] = Wt0 + (size_t)(wave*64 + j*16 + mr) * K0 + kh;

      // ---- prim contribution: 4 quarters (16x256), double-buffered DMA ----
      const long tbase = b0r*8192 + (long)t*PS;
#if HAVE_TDM
      if (wave == 0) {
        tdm_prim_tile(prim, tbase, (unsigned)(size_t)(void*)Qb[0]);
        __builtin_amdgcn_s_wait_tensorcnt(0);
      }
#else
      #pragma unroll
      for (int i = 0; i < 4; ++i) {            // 1024 float4 slots coop load
        int idx = i*256 + tid, m = idx >> 6, c4 = idx & 63;
        float4 v = *(const float4*)(prim + tbase + (long)m*8192 + c4*4);
        *(float4*)(Qb[0] + m*QROW + c4*4) = v;
      }
#endif
      __syncthreads();

      for (int q = 0; q < 4; ++q) {
        const float* cur = Qb[q & 1];
        float*       nxt = Qb[(q & 1) ^ 1];
        if (q < 3) {
#if HAVE_TDM
          if (wave == 0)
            tdm_prim_tile(prim, tbase + (q+1)*256,
                          (unsigned)(size_t)(void*)nxt);
#else
          #pragma unroll
          for (int i = 0; i < 4; ++i) {
            int idx = i*256 + tid, m = idx >> 6, c4 = idx & 63;
            float4 v = *(const float4*)(prim + tbase + (q+1)*256 +
                                        (long)m*8192 + c4*4);
            *(float4*)(nxt + m*QROW + c4*4) = v;
          }
#endif
        }
        const float* Sp = cur + mr*QROW + kh;
        const int kb = q*256;
        for (int ks = 0; ks < 256; ks += 4) {
          v2f a = *(const v2f*)(Sp + ks);
          acc[0] = wmma4(a, *(const v2f*)(bw[0] + kb + ks), acc[0]);
          acc[1] = wmma4(a, *(const v2f*)(bw[1] + kb + ks), acc[1]);
          acc[2] = wmma4(a, *(const v2f*)(bw[2] + kb + ks), acc[2]);
          acc[3] = wmma4(a, *(const v2f*)(bw[3] + kb + ks), acc[3]);
        }
#if HAVE_TDM
        if (wave == 0) __builtin_amdgcn_s_wait_tensorcnt(0);
#endif
        __syncthreads();
      }

      // ---- state (x-view) contribution: K 1024..1535 ----
      for (int ks = 0; ks < 512; ks += 4) {
        int pos = (t*F + ks + off) & 4095;          // mult of 4 -> b64 safe
        v2f a = *(const v2f*)(P + mr*SROW + pos + kh);
        acc[0] = wmma4(a, *(const v2f*)(bw[0] + 1024 + ks), acc[0]);
        acc[1] = wmma4(a, *(const v2f*)(bw[1] + 1024 + ks), acc[1]);
        acc[2] = wmma4(a, *(const v2f*)(bw[2] + 1024 + ks), acc[2]);
        acc[3] = wmma4(a, *(const v2f*)(bw[3] + 1024 + ks), acc[3]);
      }
      __syncthreads();   // all reads of state row t done -> in-place writeback
      #pragma unroll
      for (int j = 0; j < 4; ++j) {
        int n0 = wave*64 + j*16;
        float bia = eb0[col*F + n0 + mr];           // b0 + col*W0[1536,:]
        int pos = (t*F + n0 + mr + off) & 4095;
        #pragma unroll
        for (int r = 0; r < 8; ++r)
          P[(r + mhi)*SROW + pos] = fmaxf(acc[j][r] + bia, 0.f);
      }
    }
    __syncthreads();
    off = (off + 128) & 4095;                       // roll left by 128

    // ---------------- layer 1: K = 512 (state) ------------------------------
    for (int t = 0; t < NT; ++t) {
      v8f acc[4] = {vzero, vzero, vzero, vzero};
      const float* bw[4];
      #pragma unroll
      for (int j = 0; j < 4; ++j)
        bw[j] = Wt1 + (size_t)(wave*64 + j*16 + mr) * K1 + kh;

      for (int ks = 0; ks < 512; ks += 4) {
        int pos = (t*F + ks + off) & 4095;
        v2f a = *(const v2f*)(P + mr*SROW + pos + kh);
        acc[0] = wmma4(a, *(const v2f*)(bw[0] + ks), acc[0]);
        acc[1] = wmma4(a, *(const v2f*)(bw[1] + ks), acc[1]);
        acc[2] = wmma4(a, *(const v2f*)(bw[2] + ks), acc[2]);
        acc[3] = wmma4(a, *(const v2f*)(bw[3] + ks), acc[3]);
      }
      __syncthreads();
      #pragma unroll
      for (int j = 0; j < 4; ++j) {
        int n0 = wave*64 + j*16;
        float bia = eb1[col*F + n0 + mr];           // b1 + col*W1[512,:]
        int pos = (t*F + n0 + mr + off) & 4095;
        #pragma unroll
        for (int r = 0; r < 8; ++r)
          P[(r + mhi)*SROW + pos] = fmaxf(acc[j][r] + bia, 0.f);
      }
    }
    __syncthreads();
    off = (off - 128) & 4095;                       // roll right by 128
  }
  // off == 0 here: state is in logical order.

  // ---------------- output GEMM: (16 x 4096) @ (4096 x 10) + b_out ----------
  {
    float* R = lds + LDS_P;                // reuse staging LDS (needs 2560 fl)
    int m = tid >> 4, part = tid & 15;
    float s[10];
    #pragma unroll
    for (int o = 0; o < 10; ++o) s[o] = 0.f;
    for (int k = part*256; k < part*256 + 256; ++k) {
      float pv = P[m*SROW + k];
      const float* w = Wout + (size_t)k*10;
      #pragma unroll
      for (int o = 0; o < 10; ++o) s[o] += pv * w[o];
    }
    #pragma unroll
    for (int o = 0; o < 10; ++o) R[o*256 + tid] = s[o];
    __syncthreads();
    if (tid < 160) {
      int mm = tid / 10, o = tid % 10;
      float r = bout[o];
      #pragma unroll
      for (int p = 0; p < 16; ++p) r += R[o*256 + mm*16 + p];
      out[(b0r + mm)*10 + o] = r;
    }
  }
}

// ---- prep: transpose weights to k-contiguous, fold col-feature into bias ----
__global__ void prep_transpose(const float* __restrict__ W, float* __restrict__ Wt,
                               int K, int N) {
  int idx = blockIdx.x*256 + threadIdx.x;
  if (idx < K*N) { int k = idx / N, n = idx - k*N; Wt[(size_t)n*K + k] = W[(size_t)k*N + n]; }
}
__global__ void prep_bias(const float* __restrict__ W, const float* __restrict__ b,
                          float* __restrict__ eb, int K, int N) {
  int idx = blockIdx.x*256 + threadIdx.x;
  if (idx < NCOL*N) { int c = idx / N, n = idx - c*N;
                      eb[idx] = b[n] + (float)c * W[(size_t)K*N + n]; }
}

extern "C" void kernel_launch(void* const* d_in, const int* in_sizes, int n_in,
                              void* d_out, int out_size, void* d_ws, size_t ws_size,
                              hipStream_t stream) {
  (void)in_sizes; (void)n_in; (void)out_size; (void)ws_size;
  const float* prim = (const float*)d_in[0];
  const float* W0   = (const float*)d_in[1];
  const float* b0   = (const float*)d_in[2];
  const float* W1   = (const float*)d_in[3];
  const float* b1   = (const float*)d_in[4];
  const float* Wout = (const float*)d_in[5];
  const float* bout = (const float*)d_in[6];

  float* ws  = (float*)d_ws;                 // ~4.2 MB used
  float* Wt0 = ws;                           // 512 x 1536
  float* Wt1 = Wt0 + (size_t)F * K0;         // 512 x 512
  float* eb0 = Wt1 + (size_t)F * K1;         // 4 x 512
  float* eb1 = eb0 + NCOL * F;               // 4 x 512

  prep_transpose<<<(K0*F + 255)/256, 256, 0, stream>>>(W0, Wt0, K0, F);
  prep_transpose<<<(K1*F + 255)/256, 256, 0, stream>>>(W1, Wt1, K1, F);
  prep_bias<<<(NCOL*F + 255)/256, 256, 0, stream>>>(W0, b0, eb0, K0, F);
  prep_bias<<<(NCOL*F + 255)/256, 256, 0, stream>>>(W1, b1, eb1, K1, F);

  caps_main<<<NBATCH/TM, 256, 0, stream>>>(prim, Wt0, Wt1, eb0, eb1,
                                           Wout, bout, (float*)d_out);
}